// FlowAttention_88862873354608
// MI455X (gfx1250) — compile-verified
//
#include <hip/hip_runtime.h>
#include <cmath>

// ---------------- fixed problem sizes (from setup_inputs) ----------------
#define NB   64          // N
#define CH   192         // C
#define HH   112
#define WW   112
#define HWSZ (HH*WW)     // 12544, divisible by 16 and 4
#define NHW  (NB*HWSZ)   // 802816
#define SLEN 16

typedef float v2f __attribute__((ext_vector_type(2)));
typedef float v8f __attribute__((ext_vector_type(8)));

// ============ Kernel 1: 1x1 conv (channel dot product) via f32 WMMA ========
// One wave32 per 16-pixel group. A = 16(pixels) x 4(channels), B = 4 x 16
// broadcast weights (all columns identical), D accumulates over 48 K-chunks.
__global__ __launch_bounds__(256) void conv1x1_wmma(
    const float* __restrict__ in, const float* __restrict__ wc,
    float* __restrict__ x) {
  int wave = blockIdx.x * (blockDim.x >> 5) + (threadIdx.x >> 5);
  int lane = threadIdx.x & 31;
  if (wave >= NHW / 16) return;                 // uniform per wave
  size_t pixBase = (size_t)wave * 16;
  int n  = (int)(pixBase / HWSZ);
  int hw = (int)(pixBase % HWSZ);
  const float* fin = in + (size_t)n * CH * HWSZ + hw;
  int m     = lane & 15;                        // A row (pixel)
  int khalf = (lane >> 4) * 2;                  // K lanes: 0..15 -> k{0,1}; 16..31 -> k{2,3}

  v8f acc = {};
  #pragma unroll 4
  for (int c0 = 0; c0 < CH; c0 += 4) {
    v2f a, b;
    a[0] = fin[(size_t)(c0 + khalf    ) * HWSZ + m];
    a[1] = fin[(size_t)(c0 + khalf + 1) * HWSZ + m];
    b[0] = wc[c0 + khalf];
    b[1] = wc[c0 + khalf + 1];
    acc = __builtin_amdgcn_wmma_f32_16x16x4_f32(
        /*neg_a=*/false, a, /*neg_b=*/false, b,
        /*c_mod=*/(short)0, acc, /*reuse_a=*/false, /*reuse_b=*/false);
  }
  // D layout: VGPR r holds M=r (lanes 0-15) / M=r+8 (lanes 16-31); col N = lane%16.
  if ((lane & 15) == 0) {
    float* xo = x + pixBase + (lane >> 4) * 8;
    #pragma unroll
    for (int r = 0; r < 8; ++r) xo[r] = acc[r];
  }
}

// ============ Kernel 2a: five fixed 5x5 convolutions (coef 1..5) ===========
struct FarnK { float k[5][25]; };

__global__ __launch_bounds__(256) void farneback_coef(
    const float* __restrict__ x, float* __restrict__ coef, FarnK fk) {
  int idx = blockIdx.x * blockDim.x + threadIdx.x;
  if (idx >= NHW) return;
  int n = idx / HWSZ, hw = idx % HWSZ;
  int h = hw / WW, w0 = hw % WW;
  const float* f = x + (size_t)n * HWSZ;
  float c0 = 0.f, c1 = 0.f, c2 = 0.f, c3 = 0.f, c4 = 0.f;
  #pragma unroll
  for (int iu = 0; iu < 5; ++iu) {
    int hh = h + iu - 2;
    if ((unsigned)hh >= (unsigned)HH) continue;   // zero pad (SAME)
    #pragma unroll
    for (int iv = 0; iv < 5; ++iv) {
      int ww2 = w0 + iv - 2;
      if ((unsigned)ww2 >= (unsigned)WW) continue;
      float v = f[hh * WW + ww2];
      int t = iu * 5 + iv;
      c0 = fmaf(fk.k[0][t], v, c0);
      c1 = fmaf(fk.k[1][t], v, c1);
      c2 = fmaf(fk.k[2][t], v, c2);
      c3 = fmaf(fk.k[3][t], v, c3);
      c4 = fmaf(fk.k[4][t], v, c4);
    }
  }
  coef[(size_t)0 * NHW + idx] = c0;   // cu
  coef[(size_t)1 * NHW + idx] = c1;   // cv
  coef[(size_t)2 * NHW + idx] = c2;   // cuu
  coef[(size_t)3 * NHW + idx] = c3;   // cvv
  coef[(size_t)4 * NHW + idx] = c4;   // cuv
}

// ============ Kernel 2b: 2x2 flow solve + magnitude + sigmoid -> a =========
__global__ __launch_bounds__(256) void flow_mask(
    const float* __restrict__ coef, float* __restrict__ aOut) {
  int idx = blockIdx.x * blockDim.x + threadIdx.x;
  if (idx >= NHW) return;
  int n = idx / HWSZ, hw = idx % HWSZ;
  int s = n % SLEN;
  int np = (s == 0) ? n : n - 1;                 // prev frame of the pair
  int nn = (s == 0) ? n + 1 : n;                 // next frame of the pair
  size_t pp = (size_t)np * HWSZ + hw;
  size_t qq = (size_t)nn * HWSZ + hw;
  float cu_p  = coef[(size_t)0*NHW+pp], cu_n  = coef[(size_t)0*NHW+qq];
  float cv_p  = coef[(size_t)1*NHW+pp], cv_n  = coef[(size_t)1*NHW+qq];
  float cuu_p = coef[(size_t)2*NHW+pp], cuu_n = coef[(size_t)2*NHW+qq];
  float cvv_p = coef[(size_t)3*NHW+pp], cvv_n = coef[(size_t)3*NHW+qq];
  float cuv_p = coef[(size_t)4*NHW+pp], cuv_n = coef[(size_t)4*NHW+qq];

  float a11 = 0.5f  * (cuu_p + cuu_n);
  float a22 = 0.5f  * (cvv_p + cvv_n);
  float a12 = 0.25f * (cuv_p + cuv_n);
  float dbu = 0.5f  * (cu_p - cu_n);
  float dbv = 0.5f  * (cv_p - cv_n);
  float m0 = a11*a11 + a12*a12;
  float m1 = (a11 + a22) * a12;
  float m2 = a22*a22 + a12*a12;
  float m3 = a11*dbu + a12*dbv;
  float m4 = a12*dbu + a22*dbv;
  float det = m0*m2 - m1*m1;
  float inv = (fabsf(det) > 1e-9f) ? (1.0f / det) : 0.0f;
  float fu = (m2*m3 - m1*m4) * inv;
  float fv = (m0*m4 - m1*m3) * inv;
  float mag = sqrtf(fu*fu + fv*fv);
  aOut[idx] = 1.0f / (1.0f + expf(-mag)) - 0.5f;  // sigmoid - 0.5
}

// ============ Kernel 3: g = 2*a*input (float4 vectorized) ==================
__global__ __launch_bounds__(256) void gate_mul(
    const float* __restrict__ in, const float* __restrict__ aIn,
    float* __restrict__ g) {
  size_t i4 = (size_t)blockIdx.x * blockDim.x + threadIdx.x;
  const size_t total4 = (size_t)NB * CH * HWSZ / 4;
  if (i4 >= total4) return;
  size_t flat = i4 * 4;
  size_t hw = flat % HWSZ;                        // HWSZ % 4 == 0
  size_t n  = flat / ((size_t)CH * HWSZ);
  float4 in4 = *(const float4*)(in + flat);
  float4 a4  = *(const float4*)(aIn + n * HWSZ + hw);
  float4 o;
  o.x = 2.0f * a4.x * in4.x;
  o.y = 2.0f * a4.y * in4.y;
  o.z = 2.0f * a4.z * in4.z;
  o.w = 2.0f * a4.w * in4.w;
  *(float4*)(g + flat) = o;
}

// ---------------- host: fold Gaussian weights + Ginv into 5x5 kernels ------
static void compute_farneback_kernels(FarnK& fk) {
  double d[5], g[5], sum = 0.0;
  for (int i = 0; i < 5; ++i) { d[i] = i - 2; g[i] = std::exp(-d[i]*d[i] / (2.0*1.2*1.2)); sum += g[i]; }
  for (int i = 0; i < 5; ++i) g[i] /= sum;
  double phi[25][6], w[25];
  for (int iu = 0, t = 0; iu < 5; ++iu)
    for (int iv = 0; iv < 5; ++iv, ++t) {
      double u = d[iu], v = d[iv];
      phi[t][0] = 1; phi[t][1] = u; phi[t][2] = v;
      phi[t][3] = u*u; phi[t][4] = v*v; phi[t][5] = u*v;
      w[t] = g[iu] * g[iv];
    }
  double A[6][12] = {};
  for (int j = 0; j < 6; ++j) {
    for (int k = 0; k < 6; ++k) {
      double s = 0.0;
      for (int t = 0; t < 25; ++t) s += w[t] * phi[t][j] * phi[t][k];
      A[j][k] = s;
    }
    A[j][6 + j] = 1.0;
  }
  for (int col = 0; col < 6; ++col) {            // Gauss-Jordan, partial pivot
    int piv = col;
    for (int r = col + 1; r < 6; ++r) if (std::fabs(A[r][col]) > std::fabs(A[piv][col])) piv = r;
    if (piv != col) for (int k = 0; k < 12; ++k) { double t = A[col][k]; A[col][k] = A[piv][k]; A[piv][k] = t; }
    double pv = A[col][col];
    for (int k = 0; k < 12; ++k) A[col][k] /= pv;
    for (int r = 0; r < 6; ++r) if (r != col) {
      double f = A[r][col];
      for (int k = 0; k < 12; ++k) A[r][k] -= f * A[col][k];
    }
  }
  // K_j(u,v) = w(u,v) * sum_k Ginv[j+1][k] * phi_k(u,v), j = 0..4 -> coef 1..5
  for (int j = 0; j < 5; ++j)
    for (int t = 0; t < 25; ++t) {
      double s = 0.0;
      for (int k = 0; k < 6; ++k) s += A[j + 1][6 + k] * phi[t][k];
      fk.k[j][t] = (float)(w[t] * s);
    }
}

extern "C" void kernel_launch(void* const* d_in, const int* in_sizes, int n_in,
                              void* d_out, int out_size, void* d_ws, size_t ws_size,
                              hipStream_t stream) {
  (void)in_sizes; (void)n_in; (void)out_size; (void)ws_size;
  const float* input = (const float*)d_in[0];
  const float* wconv = (const float*)d_in[1];      // (1,C,1,1) -> C floats
  // d_in[2] = SL (=16), hardcoded above.

  float* aOut = (float*)d_out;                     // N*H*W floats
  float* gOut = aOut + NHW;                        // N*C*H*W floats

  float* xbuf = (float*)d_ws;                      // N*H*W       (3.2 MB)
  float* coef = xbuf + NHW;                        // 5 * N*H*W   (16 MB)

  FarnK fk;
  compute_farneback_kernels(fk);

  // k1: channel reduction via WMMA (one wave per 16 pixels)
  {
    int waves = NHW / 16;                          // 50176
    int blocks = waves / 8;                        // 8 waves of 32 per block
    conv1x1_wmma<<<blocks, 256, 0, stream>>>(input, wconv, xbuf);
  }
  // k2a: coefficient planes
  {
    int blocks = (NHW + 255) / 256;
    farneback_coef<<<blocks, 256, 0, stream>>>(xbuf, coef, fk);
  }
  // k2b: flow solve -> a
  {
    int blocks = (NHW + 255) / 256;
    flow_mask<<<blocks, 256, 0, stream>>>(coef, aOut);
  }
  // k3: g = 2*a*input
  {
    size_t total4 = (size_t)NB * CH * HWSZ / 4;
    int blocks = (int)((total4 + 255) / 256);
    gate_mul<<<blocks, 256, 0, stream>>>(input, aOut, gOut);
  }
}